// CRFLoss_19207093748192
// MI455X (gfx1250) — compile-verified
//
#include <hip/hip_runtime.h>
#include <hip/hip_bf16.h>

// Problem shape (from reference): B=64, S=512, T=32.
#define T_DIM   32
#define TT      (T_DIM * T_DIM)   // 1024 floats per (b,s) tile = 4KB
#define ROWP    36                // padded LDS row stride (144B -> 16B aligned rows)
#define END_IND (T_DIM - 1)

typedef float v2f __attribute__((ext_vector_type(2)));
typedef float v8f __attribute__((ext_vector_type(8)));
typedef int   v4i __attribute__((vector_size(16)));   // b128 payload type

// ---------------- async global -> LDS tile fetch (gfx1250 path) -------------
#if __has_builtin(__builtin_amdgcn_global_load_async_to_lds_b128) && \
    __has_builtin(__builtin_amdgcn_s_wait_asynccnt)
#define USE_ASYNC 1
#else
#define USE_ASYNC 0
#endif

#if USE_ASYNC
typedef __attribute__((address_space(1))) v4i gv4i;  // global b128
typedef __attribute__((address_space(3))) v4i lv4i;  // LDS b128
__device__ __forceinline__ void tile_fetch(const float* gsrc, float* ldst, int lane) {
  // lane l moves row l: 128B global -> LDS row (stride ROWP floats), 8 x b128.
  const float* g = gsrc + lane * T_DIM;
  float*       l = ldst + lane * ROWP;
#pragma unroll
  for (int i = 0; i < 8; ++i) {
    __builtin_amdgcn_global_load_async_to_lds_b128(
        (gv4i*)(g + i * 4), (lv4i*)(l + i * 4), 0, 0);
  }
}
#define TILE_WAIT(n) __builtin_amdgcn_s_wait_asynccnt(n)
#else
__device__ __forceinline__ void tile_fetch(const float* gsrc, float* ldst, int lane) {
  const float2* g = (const float2*)(gsrc + lane * T_DIM);
  float2*       l = (float2*)(ldst + lane * ROWP);
#pragma unroll
  for (int i = 0; i < 16; ++i) l[i] = g[i];
}
#define TILE_WAIT(n) ((void)0)
#endif

// ---------------- forward scan: one wave32 per batch element ----------------
__global__ void __launch_bounds__(32)
crf_forward_kernel(const float* __restrict__ scores,
                   const int*   __restrict__ targets,
                   const int*   __restrict__ lengths,
                   float*       __restrict__ partial,
                   int S) {
  __shared__ float tile[2][T_DIM * ROWP];  // double-buffered score tiles
  __shared__ float ebuf[T_DIM * ROWP];     // E = exp(score - rowmax)
  __shared__ float fsbuf[T_DIM];           // forward state
  __shared__ float vbuf[T_DIM];            // v = exp(fs - max fs)
  __shared__ float sbuf[T_DIM];            // (E*v) column result

  const int b    = blockIdx.x;
  const int lane = threadIdx.x;            // lane == cur tag
  const float* base = scores + (size_t)b * S * TT;
  const int len = lengths[b];              // in [1, S]

  // Stage tile 0, prefetch tile 1.
  tile_fetch(base, tile[0], lane);
  if (len > 1) {
    tile_fetch(base + TT, tile[1], lane);
    TILE_WAIT(8);                          // first 8 async ops (tile 0) done
  } else {
    TILE_WAIT(0);
  }
  __syncthreads();

  // fs0[cur] = lse_prev scores[b,0,cur,prev]
  {
    const float* row = &tile[0][lane * ROWP];
    float m = row[0];
#pragma unroll
    for (int c = 1; c < T_DIM; ++c) m = fmaxf(m, row[c]);
    float sum = 0.f;
#pragma unroll
    for (int c = 0; c < T_DIM; ++c) sum += __expf(row[c] - m);
    fsbuf[lane] = __logf(sum) + m;
  }
  float gold = 0.f;
  if (lane == 0) {
    const int t = targets[b * S];
    gold = tile[0][(t >> 5) * ROWP + (t & 31)];
  }
  __syncthreads();

  const int n16   = lane & 15;
  const int kBase = (lane < 16) ? 0 : 2;   // ISA 32-bit A/B fragment K split

  for (int s = 1; s < len; ++s) {
    const int cb = s & 1;
    TILE_WAIT(0);                          // tile[cb] resident
    __syncthreads();
    if (s + 1 < len)                       // overlap next fetch with compute
      tile_fetch(base + (size_t)(s + 1) * TT, tile[cb ^ 1], lane);

    if (lane == 0) {                       // gold gather rides on resident tile
      const int t = targets[b * S + s];
      gold += tile[cb][(t >> 5) * ROWP + (t & 31)];
    }

    // v[prev] = exp(fs[prev] - max fs)
    const float fsl = fsbuf[lane];
    float mfs = fsl;
#pragma unroll
    for (int off = 16; off; off >>= 1) mfs = fmaxf(mfs, __shfl_xor(mfs, off, 32));
    vbuf[lane] = __expf(fsl - mfs);

    // E row: lane cur owns row cur
    const float* row = &tile[cb][lane * ROWP];
    float rmax = row[0];
#pragma unroll
    for (int c = 1; c < T_DIM; ++c) rmax = fmaxf(rmax, row[c]);
    float* erow = &ebuf[lane * ROWP];
#pragma unroll
    for (int c = 0; c < T_DIM; ++c) erow[c] = __expf(row[c] - rmax);
    __syncthreads();

    // (E 32x32) x (V 32x16, col0 = v) via 2 row-blocks x 8 K-slabs of
    // V_WMMA_F32_16X16X4_F32 (full-EXEC, convergent).
#pragma unroll
    for (int i = 0; i < 2; ++i) {
      v8f acc = {0.f, 0.f, 0.f, 0.f, 0.f, 0.f, 0.f, 0.f};
      const int rowIdx = i * 16 + n16;
#pragma unroll
      for (int kb = 0; kb < 8; ++kb) {
        const int k = kb * 4 + kBase;
        v2f a = {ebuf[rowIdx * ROWP + k], ebuf[rowIdx * ROWP + k + 1]};
        const float b0 = vbuf[k], b1 = vbuf[k + 1];
        v2f bf = {(n16 == 0) ? b0 : 0.f, (n16 == 0) ? b1 : 0.f};
        acc = __builtin_amdgcn_wmma_f32_16x16x4_f32(
            false, a, false, bf, (short)0, acc, false, false);
      }
      // D column 0: lane 0 holds rows 0..7 of the block, lane 16 rows 8..15.
      if (lane == 0) {
#pragma unroll
        for (int r = 0; r < 8; ++r) sbuf[i * 16 + r] = acc[r];
      } else if (lane == 16) {
#pragma unroll
        for (int r = 0; r < 8; ++r) sbuf[i * 16 + 8 + r] = acc[r];
      }
    }
    __syncthreads();
    const float newfs = __logf(sbuf[lane]) + rmax + mfs;
    __syncthreads();
    fsbuf[lane] = newfs;
    __syncthreads();
  }

  if (lane == 0) partial[b] = fsbuf[END_IND] - gold;
}

// ---------------- deterministic fixed-order reduction -----------------------
__global__ void crf_reduce_kernel(const float* __restrict__ partial,
                                  float* __restrict__ out, int n) {
  if (blockIdx.x == 0 && threadIdx.x == 0) {
    float s = 0.f;
    for (int i = 0; i < n; ++i) s += partial[i];
    out[0] = s;
  }
}

extern "C" void kernel_launch(void* const* d_in, const int* in_sizes, int n_in,
                              void* d_out, int out_size, void* d_ws, size_t ws_size,
                              hipStream_t stream) {
  const float* scores  = (const float*)d_in[0];  // [B,S,T,T] fp32
  const int*   targets = (const int*)d_in[1];    // [B,S] int32
  const int*   lengths = (const int*)d_in[2];    // [B]   int32
  float* out     = (float*)d_out;
  float* partial = (float*)d_ws;                 // B floats of scratch

  const int B = in_sizes[2];
  const int S = in_sizes[1] / B;

  crf_forward_kernel<<<B, 32, 0, stream>>>(scores, targets, lengths, partial, S);
  crf_reduce_kernel<<<1, 32, 0, stream>>>(partial, out, B);
}